// CentroidDistance_61847529062760
// MI455X (gfx1250) — compile-verified
//
#include <hip/hip_runtime.h>
#include <math.h>
#include <stdint.h>

typedef __attribute__((ext_vector_type(2))) float v2f;
typedef __attribute__((ext_vector_type(8))) float v8f;
typedef __attribute__((ext_vector_type(4))) unsigned v4u;
typedef __attribute__((ext_vector_type(8))) unsigned v8u;

#define EPSF 1e-5f
constexpr int D = 128;           // feature dim
constexpr int K = 128;           // centroids
constexpr int TILE_M = 128;      // rows per block
constexpr int LSTRIDE = 132;     // padded LDS row stride (floats) == TDM pad 128+4 DWORDs

// workspace layout (floats):
constexpr int WS_C2   = 16384;   // c2[K]
constexpr int WS_COL  = 16512;   // colsum[K]  (zeroed each launch)
constexpr int WS_MSUM = 16640;   // masksum

__global__ __launch_bounds__(256) void init_kernel(const float* __restrict__ cw,
                                                   const float* __restrict__ mask,
                                                   int N, float* __restrict__ ws) {
    const int tid = threadIdx.x;
    if (tid < K) {
        ws[WS_COL + tid] = 0.0f;
        const float* row = cw + tid * D;
        float ss = 0.0f;
        for (int d = 0; d < D; ++d) { float v = row[d]; ss += v * v; }
        float nrm = sqrtf(ss);
        float ncl = fmaxf(nrm, EPSF);
        float t   = tanhf(fminf(ncl, 15.0f));
        float scale = t / ncl;
        for (int d = 0; d < D; ++d) ws[tid * D + d] = row[d] * scale;
        ws[WS_C2 + tid] = scale * scale * ss;
    }
    __shared__ float red[256];
    float s = 0.0f;
    for (int i = tid; i < N; i += 256) s += mask[i];
    red[tid] = s;
    __syncthreads();
    for (int off = 128; off > 0; off >>= 1) {
        if (tid < off) red[tid] += red[tid + off];
        __syncthreads();
    }
    if (tid == 0) ws[WS_MSUM] = red[0];
}

// Build TDM D# groups for a 2D f32 tile load: rows x 128 floats, row stride 128,
// LDS-padded to 132 DWORDs per row (pad_interval=6 -> every 128 DWORDs,
// pad_amount=3 -> 4 DWORDs). OOB rows (beyond tensor_dim1) read as zero.
__device__ __forceinline__ void tdm_load_tile(const float* gsrc, unsigned lds_off,
                                              unsigned tensor_rows) {
    const unsigned long long ga = (unsigned long long)(uintptr_t)gsrc;
    v4u g0;
    g0[0] = 1u;                                            // count=1, user mode
    g0[1] = lds_off;                                       // lds_addr (bytes)
    g0[2] = (unsigned)(ga & 0xFFFFFFFFull);                // global_addr[31:0]
    g0[3] = ((unsigned)((ga >> 32) & 0x01FFFFFFull))       // global_addr[56:32]
          | 0x80000000u;                                   // type=2 ("image")
    v8u g1;
    g1[0] = (2u << 16)      // data_size = 4B
          | (1u << 20)      // pad_enable
          | (6u << 22)      // pad_interval: every 128 DWORDs
          | (3u << 25);     // pad_amount: 4 DWORDs
    g1[1] = (128u & 0xFFFFu) << 16;                        // tensor_dim0 = 128 (low16)
    g1[2] = (tensor_rows & 0xFFFFu) << 16;                 // dim0 hi16=0 | tensor_dim1 low16
    g1[3] = ((tensor_rows >> 16) & 0xFFFFu)                // tensor_dim1 high16
          | (128u << 16);                                  // tile_dim0 = 128
    g1[4] = 128u;                                          // tile_dim1 = 128, tile_dim2 = 0
    g1[5] = 128u;                                          // tensor_dim0_stride[31:0] = 128
    g1[6] = 0u;                                            // stride hi | dim1_stride lo
    g1[7] = 0u;
    asm volatile("tensor_load_to_lds %0, %1" :: "s"(g0), "s"(g1) : "memory");
}

__global__ __launch_bounds__(256) void dist_kernel(const float* __restrict__ x,
                                                   const float* __restrict__ mask,
                                                   const float* __restrict__ ws,
                                                   float* __restrict__ out,
                                                   float* __restrict__ colsum_g,
                                                   int N) {
    __shared__ float xs[TILE_M * LSTRIDE];   // x tile, TDM-padded
    __shared__ float cs[K * LSTRIDE];        // centroid_repr, TDM-padded
    __shared__ float c2s[K];
    __shared__ float colacc[K];

    const int tid  = threadIdx.x;
    const int wave = tid >> 5;
    const int lane = tid & 31;
    const int half = lane >> 4;
    const int l16  = lane & 15;
    const int base = blockIdx.x * TILE_M;

    // ---- async tile staging via Tensor Data Mover (one issuing wave) ----
    if (tid < 32) {
        tdm_load_tile(x + (size_t)base * D,
                      (unsigned)(uintptr_t)(void*)xs,
                      (unsigned)(N - base));               // tail rows -> zeros
        tdm_load_tile(ws,                                  // centroid_repr at ws[0]
                      (unsigned)(uintptr_t)(void*)cs, (unsigned)K);
    }
    if (tid < K) { c2s[tid] = ws[WS_C2 + tid]; colacc[tid] = 0.0f; }
    if (tid < 32) __builtin_amdgcn_s_wait_tensorcnt(0);
    __syncthreads();

    // ---- GEMM: 16x16x4 f32 WMMA, 32 K-steps, 8 N-tiles per wave ----
    v8f acc[8] = {};
    float xsq = 0.0f;
    const int arow = wave * 16 + l16;
    for (int kk = 0; kk < 32; ++kk) {
        const int koff = kk * 4 + half * 2;
        v2f a = *(const v2f*)&xs[arow * LSTRIDE + koff];
        xsq += a.x * a.x + a.y * a.y;
#pragma unroll
        for (int nt = 0; nt < 8; ++nt) {
            v2f b = *(const v2f*)&cs[(nt * 16 + l16) * LSTRIDE + koff];
            acc[nt] = __builtin_amdgcn_wmma_f32_16x16x4_f32(
                false, a, false, b, (short)0, acc[nt], false, false);
        }
    }

    float xsqFull = xsq + __shfl_xor(xsq, 16, 32);

    const int growLane = base + wave * 16 + l16;
    float mrow = (growLane < N) ? mask[growLane] : 0.0f;

    // ---- epilogue: poincare distance + outputs + column sums ----
#pragma unroll
    for (int nt = 0; nt < 8; ++nt) {
        const int col = nt * 16 + l16;
        const float c2v = c2s[col];
        float cpart = 0.0f;
#pragma unroll
        for (int i = 0; i < 8; ++i) {
            const int rloc = i + half * 8;
            const float x2r = __shfl(xsqFull, rloc, 32);
            const float mr  = __shfl(mrow,    rloc, 32);
            const float dot = acc[nt][i];
            float sq  = fmaxf(x2r + c2v - 2.0f * dot, 0.0f);
            float den = fmaxf((1.0f - x2r) * (1.0f - c2v), EPSF);
            float arg = fmaxf(1.0f + 2.0f * sq / den, 1.0f + 1e-7f);
            float dist = logf(arg + sqrtf(arg * arg - 1.0f));  // arccosh
            float dm = dist * mr;
            cpart += dm;
            const int grow = base + wave * 16 + rloc;
            if (grow < N) out[K + (size_t)grow * K + col] = dm;
        }
        cpart += __shfl_xor(cpart, 16, 32);
        if (half == 0) atomicAdd(&colacc[col], cpart);
    }
    __syncthreads();
    if (tid < K) atomicAdd(&colsum_g[tid], colacc[tid]);
}

__global__ void finalize_kernel(const float* __restrict__ ws, float* __restrict__ out) {
    const int t = threadIdx.x;
    if (t < K) out[t] = ws[WS_COL + t] / ws[WS_MSUM];
}

extern "C" void kernel_launch(void* const* d_in, const int* in_sizes, int n_in,
                              void* d_out, int out_size, void* d_ws, size_t ws_size,
                              hipStream_t stream) {
    const float* node = (const float*)d_in[0];   // [N, D]
    const float* mask = (const float*)d_in[1];   // [N, 1]
    const float* cw   = (const float*)d_in[2];   // [K, D]
    float* out = (float*)d_out;                  // [K] graph ++ [N*K] node dists
    float* ws  = (float*)d_ws;
    const int N = in_sizes[0] / D;

    init_kernel<<<1, 256, 0, stream>>>(cw, mask, N, ws);
    const int nb = (N + TILE_M - 1) / TILE_M;
    dist_kernel<<<nb, 256, 0, stream>>>(node, mask, ws, out, ws + WS_COL, N);
    finalize_kernel<<<1, 128, 0, stream>>>(ws, out);
}